// PoseVocab_15710990369687
// MI455X (gfx1250) — compile-verified
//
#include <hip/hip_runtime.h>

// Problem constants (from the reference)
#define J_ 24
#define P_ 8
#define L_ 128
#define C_ 8
#define N_ 65536
#define JSTRIDE   (P_ * L_ * L_ * C_)   // 1048576 floats between j slices
#define PIDSTRIDE (L_ * L_ * C_)        // 131072 floats between pid slices

// Probe for the gfx1250 async global->LDS builtin (per-pass: host pass lacks it,
// device pass for gfx1250 should have it; falls back to plain loads otherwise).
#if defined(__has_builtin)
#if __has_builtin(__builtin_amdgcn_global_load_async_to_lds_b128)
#define HAVE_ASYNC_LDS 1
#endif
#endif

#if defined(HAVE_ASYNC_LDS)
// Toolchain-probed parameter type: 'int __attribute__((vector_size(16))) __device__ *'
typedef int pv_v4i __attribute__((ext_vector_type(4)));
typedef __attribute__((address_space(1))) pv_v4i* pv_gv4p;  // global int4*
typedef __attribute__((address_space(3))) pv_v4i* pv_lv4p;  // LDS int4*
#endif

// ---------------------------------------------------------------------------
// Kernel 1: partial sums of query_points -> ws[0..2] (raw sums; mean taken in
// the main kernel). 64 blocks x 256 threads, each block covers 1024 points.
// Demonstrates the CDNA5 async-to-LDS path (ASYNCcnt) where available.
// ---------------------------------------------------------------------------
__global__ __launch_bounds__(256) void pv_sum_kernel(const float* __restrict__ q,
                                                     float* __restrict__ ws) {
  __shared__ float stage[768];      // 256 points * 3 floats
  __shared__ float red[3 * 256];
  const int tid = threadIdx.x;
  float sx = 0.f, sy = 0.f, sz = 0.f;

  for (int chunk = 0; chunk < 4; ++chunk) {
    const int pbase = blockIdx.x * 1024 + chunk * 256;
#if defined(HAVE_ASYNC_LDS)
    // 256 points * 12B = 3072B = 192 x 16B async transfers (lanes 0..191).
    if (tid < 192) {
      const float* g = q + (size_t)pbase * 3 + (size_t)tid * 4;
      __builtin_amdgcn_global_load_async_to_lds_b128(
          (pv_gv4p)g,
          (pv_lv4p)(&stage[tid * 4]),
          /*offset=*/0, /*cpol=*/0);
    }
#if __has_builtin(__builtin_amdgcn_s_wait_asynccnt)
    __builtin_amdgcn_s_wait_asynccnt(0);
#else
    asm volatile("s_wait_asynccnt 0" ::: "memory");
#endif
    __syncthreads();
    sx += stage[tid * 3 + 0];
    sy += stage[tid * 3 + 1];
    sz += stage[tid * 3 + 2];
    __syncthreads();
#else
    const int p = pbase + tid;
    sx += q[3 * p + 0];
    sy += q[3 * p + 1];
    sz += q[3 * p + 2];
#endif
  }

  red[tid] = sx; red[256 + tid] = sy; red[512 + tid] = sz;
  __syncthreads();
  for (int s = 128; s > 0; s >>= 1) {
    if (tid < s) {
      red[tid]       += red[tid + s];
      red[256 + tid] += red[256 + tid + s];
      red[512 + tid] += red[512 + tid + s];
    }
    __syncthreads();
  }
  if (tid == 0) {
    unsafeAtomicAdd(&ws[0], red[0]);
    unsafeAtomicAdd(&ws[1], red[256]);
    unsafeAtomicAdd(&ws[2], red[512]);
  }
}

// ---------------------------------------------------------------------------
// Kernel 2: one thread per sampled point m. For each of the 3 tensors:
// compute the bilinear corners once (shared by all 192 channels), then per
// j-slice load 4 corners x 8 contiguous floats (two float4 each), combine,
// and scatter w[n',j] * s into out via float atomics, where the consumer
// index follows the torch-.view scrambling:
//   linear = ch*65536 + m ; n' = linear/192 ; j = (linear%192)/8 ; c = %8
// maintained incrementally (+65536 == +341*192 + 64).
// ---------------------------------------------------------------------------
__global__ __launch_bounds__(256) void pv_main_kernel(
    const float* __restrict__ fx, const float* __restrict__ fy,
    const float* __restrict__ fz, const float* __restrict__ q,
    const float* __restrict__ scale, const float* __restrict__ skin,
    const int* __restrict__ idp, const float* __restrict__ sums,
    float* __restrict__ out) {
  const unsigned m = blockIdx.x * 256u + threadIdx.x;
  const int pid = idp[0];
  const float inv = 1.0f / (float)N_;
  const float mx = sums[0] * inv, my = sums[1] * inv, mz = sums[2] * inv;
  const float x = 2.0f * (q[3 * m + 0] - mx) / scale[0];
  const float y = 2.0f * (q[3 * m + 1] - my) / scale[1];
  const float z = 2.0f * (q[3 * m + 2] - mz) / scale[2];

  // grid per tensor: t0 uses (gx=y, gy=x); t1 (gx=z, gy=y); t2 (gx=x, gy=z)
  const float gxs[3] = {y, z, x};
  const float gys[3] = {x, y, z};
  const float* feats[3] = {fx, fy, fz};

  const unsigned n0 = m / 192u;
  const unsigned r0 = m - n0 * 192u;

#pragma unroll
  for (int t = 0; t < 3; ++t) {
    const float ix = (gxs[t] + 1.0f) * 0.5f * (float)(L_ - 1);
    const float iy = (gys[t] + 1.0f) * 0.5f * (float)(L_ - 1);
    const float ix0f = floorf(ix);
    const float iy0f = floorf(iy);
    const float fxw = ix - ix0f;   // weight toward ix1
    const float fyw = iy - iy0f;   // weight toward iy1
    const int ix0 = (int)ix0f;
    const int iy0 = (int)iy0f;
    const int ix0c = min(max(ix0, 0), L_ - 1);
    const int ix1c = min(max(ix0 + 1, 0), L_ - 1);
    const int iy0c = min(max(iy0, 0), L_ - 1);
    const int iy1c = min(max(iy0 + 1, 0), L_ - 1);
    const float w00 = (1.f - fxw) * (1.f - fyw);  // (iy0, ix0)  nw
    const float w01 = fxw * (1.f - fyw);          // (iy0, ix1)  ne
    const float w10 = (1.f - fxw) * fyw;          // (iy1, ix0)  sw
    const float w11 = fxw * fyw;                  // (iy1, ix1)  se
    const int o00 = (iy0c * L_ + ix0c) * C_;
    const int o01 = (iy0c * L_ + ix1c) * C_;
    const int o10 = (iy1c * L_ + ix0c) * C_;
    const int o11 = (iy1c * L_ + ix1c) * C_;
    const float* base = feats[t] + (size_t)pid * PIDSTRIDE;

    unsigned np = n0;
    unsigned r = r0;
    for (int j = 0; j < J_; ++j) {
      const float* p = base + (size_t)j * JSTRIDE;
      if (j + 1 < J_) {
        __builtin_prefetch(p + JSTRIDE + o00, 0, 1);  // -> global_prefetch_b8
      }
      const float4 a00 = *(const float4*)(p + o00);
      const float4 b00 = *(const float4*)(p + o00 + 4);
      const float4 a01 = *(const float4*)(p + o01);
      const float4 b01 = *(const float4*)(p + o01 + 4);
      const float4 a10 = *(const float4*)(p + o10);
      const float4 b10 = *(const float4*)(p + o10 + 4);
      const float4 a11 = *(const float4*)(p + o11);
      const float4 b11 = *(const float4*)(p + o11 + 4);

      float sv[8];
      sv[0] = w00 * a00.x + w01 * a01.x + w10 * a10.x + w11 * a11.x;
      sv[1] = w00 * a00.y + w01 * a01.y + w10 * a10.y + w11 * a11.y;
      sv[2] = w00 * a00.z + w01 * a01.z + w10 * a10.z + w11 * a11.z;
      sv[3] = w00 * a00.w + w01 * a01.w + w10 * a10.w + w11 * a11.w;
      sv[4] = w00 * b00.x + w01 * b01.x + w10 * b10.x + w11 * b11.x;
      sv[5] = w00 * b00.y + w01 * b01.y + w10 * b10.y + w11 * b11.y;
      sv[6] = w00 * b00.z + w01 * b01.z + w10 * b10.z + w11 * b11.z;
      sv[7] = w00 * b00.w + w01 * b01.w + w10 * b10.w + w11 * b11.w;

#pragma unroll
      for (int cim = 0; cim < 8; ++cim) {
        const unsigned jj = r >> 3;
        const unsigned cc = r & 7u;
        const float wt = skin[np * 24u + jj];
        unsafeAtomicAdd(&out[np * 24u + (unsigned)t * 8u + cc], wt * sv[cim]);
        // advance linear by 65536 = 341*192 + 64
        r += 64u;
        const unsigned carry = (r >= 192u) ? 1u : 0u;
        r -= 192u * carry;
        np += 341u + carry;
      }
    }
  }
}

extern "C" void kernel_launch(void* const* d_in, const int* in_sizes, int n_in,
                              void* d_out, int out_size, void* d_ws, size_t ws_size,
                              hipStream_t stream) {
  const float* fx = (const float*)d_in[0];
  const float* fy = (const float*)d_in[1];
  const float* fz = (const float*)d_in[2];
  const float* q = (const float*)d_in[3];
  const float* sc = (const float*)d_in[4];
  const float* sw = (const float*)d_in[5];
  const int* idp = (const int*)d_in[6];
  float* out = (float*)d_out;
  float* sums = (float*)d_ws;  // 3 floats of scratch

  (void)hipMemsetAsync(d_ws, 0, 3 * sizeof(float), stream);
  (void)hipMemsetAsync(d_out, 0, (size_t)out_size * sizeof(float), stream);
  pv_sum_kernel<<<N_ / 1024, 256, 0, stream>>>(q, sums);
  pv_main_kernel<<<N_ / 256, 256, 0, stream>>>(fx, fy, fz, q, sc, sw, idp, sums, out);
}